// Spectral_attack_by_frame_52432960749887
// MI455X (gfx1250) — compile-verified
//
#include <hip/hip_runtime.h>

// Spectral attack + per-(s,f) mean/cov statistics for MI455X (gfx1250).
// One wave32 per (stream, s, f). All three matmul stages use
// V_WMMA_F32_16X16X4_F32 (f32 keeps reference precision); exp/log run on VALU.
// DCT/IDCT B-operands are precomputed once into d_ws by a setup kernel.

#define MFCC 13
#define SPEC 24
#define S_DIM 4
#define F_DIM 1128
#define T_DIM 250
#define SF (S_DIM * F_DIM)

typedef float v2f __attribute__((ext_vector_type(2)));
typedef float v4f __attribute__((ext_vector_type(4)));
typedef float v8f __attribute__((ext_vector_type(8)));

// Per-lane B-operand values in the assumed 4x16 f32 B layout:
//   VGPR j, lane-half lh -> K = 4c + 2*lh + j, N = lane&15
// f[0..7]  = B1 tile0 (IDCT^T, n = lm)      c-major, j inner
// f[8..15] = B1 tile1 (IDCT^T, n = 16+lm)
// f[16..27]= B2       (DCT^T,  m = lm)
__device__ __forceinline__ void compute_btab(int l, float* f) {
    const float PI_F = 3.14159265358979323846f;
    const int lh = l >> 4, lm = l & 15;
    int idx = 0;
    for (int nt = 0; nt < 2; ++nt)
        for (int c = 0; c < 4; ++c)
            for (int j = 0; j < 2; ++j) {
                int k = 4 * c + 2 * lh + j;
                int n = 16 * nt + lm;
                float v = 0.0f;
                if (k < MFCC && n < SPEC)
                    v = (k == 0) ? (1.0f / (2.0f * SPEC))
                                 : (cosf(PI_F * (2 * n + 1) * k / (2.0f * SPEC)) / SPEC);
                f[idx++] = v;
            }
    for (int c = 0; c < 6; ++c)
        for (int j = 0; j < 2; ++j) {
            int n = 4 * c + 2 * lh + j;
            f[idx++] = (lm < MFCC) ? 2.0f * cosf(PI_F * (2 * n + 1) * lm / (2.0f * SPEC))
                                   : 0.0f;
        }
}

__global__ __launch_bounds__(32) void btab_setup_kernel(v4f* __restrict__ btab) {
    float f[28];
    compute_btab(threadIdx.x, f);
    for (int k = 0; k < 7; ++k)
        btab[k * 32 + threadIdx.x] = (v4f){f[4 * k], f[4 * k + 1], f[4 * k + 2], f[4 * k + 3]};
}

__global__ __launch_bounds__(32) void spectral_attack_stats_kernel(
    const float* __restrict__ p_vects, const float* __restrict__ q_vects,
    const float* __restrict__ p_mask,  const float* __restrict__ q_mask,
    const float* __restrict__ npm_arr, const float* __restrict__ noise_root,
    float* __restrict__ out, const v4f* __restrict__ btab)
{
    __shared__ alignas(16) float Xs[224];       // flat 16x13 X tile (+pad, stays finite)
    __shared__ alignas(16) float Ms[224];       // flat 16x13 mask tile
    __shared__ alignas(16) float Zs[16 * 34];   // rows=frames, cols=spectral n (padded stride)
    __shared__ alignas(16) float Ot[16 * 18];   // masked Out^T: rows=mfcc m, cols=frames
    __shared__ float red[32];
    __shared__ float means_s[16];
    __shared__ float len_s;

    const int bid = blockIdx.x;
    const int stream = bid / SF;                // 0 = p, 1 = q
    const int sf = bid - stream * SF;

    const float* __restrict__ x   = stream ? q_vects : p_vects;
    const float* __restrict__ msk = stream ? q_mask  : p_mask;
    const size_t base = (size_t)sf * (T_DIM * MFCC);

    const int l  = threadIdx.x;
    const int lh = l >> 4;
    const int lm = l & 15;

    // ---- B-operand constants: table load (7 coalesced b128) or in-kernel fallback
    float f[28];
    if (btab) {
#pragma unroll
        for (int k = 0; k < 7; ++k) {
            v4f t = btab[k * 32 + l];
            f[4 * k] = t.x; f[4 * k + 1] = t.y; f[4 * k + 2] = t.z; f[4 * k + 3] = t.w;
        }
    } else {
        compute_btab(l, f);
    }
#define B1(nt, c, j) f[(nt) * 8 + (c) * 2 + (j)]
#define B2(c, j)     f[16 + (c) * 2 + (j)]

    const float noise0 = expf(noise_root[lm]);
    const float noise1 = (lm < SPEC - 16) ? expf(noise_root[16 + lm]) : 0.0f;

    // ---- lengths = sum_t mask[s,f,t,0]
    float lp = 0.0f;
    for (int t = l; t < T_DIM; t += 32) lp += msk[base + (size_t)t * MFCC];
    red[l] = lp;
    // zero pad regions of Xs/Ms (stay finite/zero for all full tiles)
    for (int i = l; i < 224; i += 32) { Xs[i] = 0.0f; Ms[i] = 0.0f; }
    __syncthreads();
    if (l == 0) {
        float s = 0.0f;
        for (int i = 0; i < 32; ++i) s += red[i];
        len_s = s;
    }
    __syncthreads();
    const float invlen = 1.0f / len_s;

    v8f m2acc = {};        // C/D layout: row = r + 8*lh, col = lm
    float accMean = 0.0f;  // column partial (col = lm, this half's frames)

    auto process_tile = [&]() {
        // matmul1: Y[16x24] = X[16x13] @ IDCT^T  (8 WMMAs); B1 zero for k>=13
        v8f y0 = {}, y1 = {};
#pragma unroll
        for (int c = 0; c < 4; ++c) {
            const int off = lm * 13 + 4 * c + 2 * lh;
            v2f a = { Xs[off], Xs[off + 1] };
            v2f bb0 = { B1(0, c, 0), B1(0, c, 1) };
            v2f bb1 = { B1(1, c, 0), B1(1, c, 1) };
            y0 = __builtin_amdgcn_wmma_f32_16x16x4_f32(false, a, false, bb0, (short)0, y0, false, false);
            y1 = __builtin_amdgcn_wmma_f32_16x16x4_f32(false, a, false, bb1, (short)0, y1, false, false);
        }
        // elementwise log(exp(y)+noise), stage Z frame-major
#pragma unroll
        for (int r = 0; r < 8; ++r) {
            const int fr = r + 8 * lh;
            Zs[fr * 34 + lm]      = logf(expf(y0[r]) + noise0);
            Zs[fr * 34 + 16 + lm] = logf(expf(y1[r]) + noise1);
        }
        __syncthreads();
        // matmul2: Out[16x13] = Z[16x24] @ DCT^T  (6 WMMAs over n-chunks)
        v8f o = {};
#pragma unroll
        for (int c = 0; c < 6; ++c) {
            v2f a  = *(const v2f*)&Zs[lm * 34 + 4 * c + 2 * lh];
            v2f bb = { B2(c, 0), B2(c, 1) };
            o = __builtin_amdgcn_wmma_f32_16x16x4_f32(false, a, false, bb, (short)0, o, false, false);
        }
        // mask (from LDS), mean partials, stage Out^T
#pragma unroll
        for (int r = 0; r < 8; ++r) {
            const int fr = r + 8 * lh;
            const float mv = (lm < MFCC) ? Ms[fr * 13 + lm] : 0.0f;
            const float ov = o[r] * mv;
            accMean += ov;
            Ot[lm * 18 + fr] = ov;
        }
        __syncthreads();
        // M2 += Out^T @ Out : A and B per-lane values identical (4 WMMAs)
#pragma unroll
        for (int c = 0; c < 4; ++c) {
            v2f p = *(const v2f*)&Ot[lm * 18 + 4 * c + 2 * lh];
            m2acc = __builtin_amdgcn_wmma_f32_16x16x4_f32(false, p, false, p, (short)0, m2acc, false, false);
        }
    };

    const v2f* __restrict__ xg2 = (const v2f*)(x + base);    // 8B-aligned (13000*sf, 832*tile)
    const v2f* __restrict__ mg2 = (const v2f*)(msk + base);
    v2f* Xs2 = (v2f*)Xs;
    v2f* Ms2 = (v2f*)Ms;

    // 15 full tiles (frames 0..239): guard-free wide staging
#pragma unroll 1
    for (int tile = 0; tile < 15; ++tile) {
        const int e0 = tile * 104;   // float2 elements per tile = 208/2
        __syncthreads();
#pragma unroll
        for (int i = 0; i < 3; ++i) {
            Xs2[l + 32 * i] = xg2[e0 + l + 32 * i];
            Ms2[l + 32 * i] = mg2[e0 + l + 32 * i];
        }
        if (l < 8) {
            Xs2[l + 96] = xg2[e0 + l + 96];
            Ms2[l + 96] = mg2[e0 + l + 96];
        }
        __syncthreads();
        process_tile();
    }

    // tail tile: frames 240..249 -> 130 floats = 65 float2; rest zero (mask=0 kills them)
    __syncthreads();
    for (int i = l; i < 224; i += 32) { Xs[i] = 0.0f; Ms[i] = 0.0f; }
    __syncthreads();
    {
        const int e0 = 15 * 104;
#pragma unroll
        for (int i = 0; i < 3; ++i) {
            const int idx = l + 32 * i;
            if (idx < 65) {
                Xs2[idx] = xg2[e0 + idx];
                Ms2[idx] = mg2[e0 + idx];
            }
        }
        __syncthreads();
        process_tile();
    }

    // ---- finalize: means, covariances
    red[l] = accMean;
    __syncthreads();
    if (l < 16) {
        const float mn = (red[l] + red[l + 16]) * invlen;
        means_s[l] = (l < MFCC) ? mn : 0.0f;
    }
    __syncthreads();

    const int SFM = SF * MFCC;
    const int SFC = SF * MFCC * MFCC;
    float* meansOut = out + (stream ? (SFM + SFC) : 0);
    float* covsOut  = out + (stream ? (2 * SFM + SFC) : SFM);

    if (l < MFCC) meansOut[(size_t)sf * MFCC + l] = means_s[l];

    const float npm = npm_arr[sf];
    const float meanCol = means_s[lm];
#pragma unroll
    for (int r = 0; r < 8; ++r) {
        const int row = r + 8 * lh;
        if (row < MFCC && lm < MFCC) {
            const float cov = m2acc[r] * invlen - means_s[row] * meanCol;
            const float e = (row == lm) ? 1.0f : 0.0f;
            const float val = (cov - e) * npm + e + 0.001f * e;
            covsOut[(size_t)sf * (MFCC * MFCC) + row * MFCC + lm] = val;
        }
    }
}

extern "C" void kernel_launch(void* const* d_in, const int* in_sizes, int n_in,
                              void* d_out, int out_size, void* d_ws, size_t ws_size,
                              hipStream_t stream) {
    (void)in_sizes; (void)n_in; (void)out_size;
    const float* p_vects = (const float*)d_in[0];
    const float* q_vects = (const float*)d_in[1];
    const float* p_mask  = (const float*)d_in[2];
    const float* q_mask  = (const float*)d_in[3];
    const float* npm     = (const float*)d_in[4];
    const float* nroot   = (const float*)d_in[5];
    float* out = (float*)d_out;

    v4f* btab = nullptr;
    if (ws_size >= 7 * 32 * sizeof(v4f)) {
        btab = (v4f*)d_ws;
        btab_setup_kernel<<<1, 32, 0, stream>>>(btab);
    }

    dim3 grid(2 * SF);   // 9024 blocks, one wave32 each
    spectral_attack_stats_kernel<<<grid, 32, 0, stream>>>(
        p_vects, q_vects, p_mask, q_mask, npm, nroot, out, btab);
}